// LookupLanguageModel_69398081568858
// MI455X (gfx1250) — compile-verified
//
#include <hip/hip_runtime.h>

// ---- problem constants (match reference) ----
#define V_   32768
#define C_   32
#define B_   32
#define U_   (V_ + 1)          // first bigram node index
#define X_   (V_ * C_ + 1)     // pointers length
#define G_   V_                // unigram count
// backoff weight for unigram node h lives at logs[X_ + G_ + h]

#define TILE_V 1024            // unigram tile per block (4 KB)
#define TPB    256             // 8 waves of 32 (wave32)

typedef int v4i __attribute__((ext_vector_type(4)));

// ---------------------------------------------------------------------------
// Kernel 1: out[b, v] = logs[X+G+h_b] + logs[v]   (dense base fill, 4 MB)
// Stages the 4 KB logs[v] tile into LDS via gfx1250 async global->LDS copy
// (ASYNCcnt path), then broadcast-adds the per-row backoff with float4 stores.
// ---------------------------------------------------------------------------
__global__ __launch_bounds__(TPB) void lm_base_fill(
    const int* __restrict__ hist, const int* __restrict__ idxp,
    const float* __restrict__ logs, float* __restrict__ out)
{
    __shared__ __align__(16) float s_uni[TILE_V];

    const int tid = threadIdx.x;
    const int vt  = blockIdx.x * TILE_V;   // vocab tile base
    const int b   = blockIdx.y;            // batch row

    const int   idx     = idxp[0];
    const int   h       = hist[(idx - 1) * B_ + b];        // uniform per block
    const float backoff = logs[X_ + G_ + h];               // scalar load

    // Async-copy logs[vt .. vt+TILE_V) into LDS: 16 bytes per lane.
    {
        const float* src = logs + vt + tid * 4;
#if __has_builtin(__builtin_amdgcn_global_load_async_to_lds_b128)
        __builtin_amdgcn_global_load_async_to_lds_b128(
            (v4i*)src, (v4i*)&s_uni[tid * 4], 0, 0);
#else
        unsigned lds_off = (unsigned)(size_t)&s_uni[tid * 4];
        asm volatile("global_load_async_to_lds_b128 %0, %1, off"
                     :: "v"(lds_off), "v"(src) : "memory");
#endif
#if __has_builtin(__builtin_amdgcn_s_wait_asynccnt)
        __builtin_amdgcn_s_wait_asynccnt(0);
#else
        asm volatile("s_wait_asynccnt 0" ::: "memory");
#endif
    }
    __syncthreads();

    float4 u = *reinterpret_cast<const float4*>(&s_uni[tid * 4]);
    float4 r;
    r.x = backoff + u.x;
    r.y = backoff + u.y;
    r.z = backoff + u.z;
    r.w = backoff + u.w;
    *reinterpret_cast<float4*>(&out[(size_t)b * V_ + vt + tid * 4]) = r;
}

// ---------------------------------------------------------------------------
// Kernel 2: scatter the B*C = 1024 bigram hits over the base fill.
// Exact reference semantics (sum over duplicate matching children):
//   phase A: matched slots <- 0, fence+barrier, phase B: atomicAdd child_lp.
// ---------------------------------------------------------------------------
__global__ __launch_bounds__(B_ * C_) void lm_scatter(
    const int* __restrict__ hist, const int* __restrict__ idxp,
    const int* __restrict__ pointers, const int* __restrict__ ids,
    const float* __restrict__ logs, float* __restrict__ out)
{
    const int tid = threadIdx.x;
    const int b   = tid / C_;
    const int j   = tid % C_;

    const int idx = idxp[0];
    const int h   = hist[(idx - 1) * B_ + b];
    const int off = pointers[h];
    const int nch = pointers[h + 1] - off + 1;

    const bool valid = (j < nch);
    int   tok  = 0;
    float lp   = 0.0f;
    if (valid) {
        const int node = h + off + j;          // bigram node id
        tok = ids[node - U_];
        lp  = logs[node];
        out[(size_t)b * V_ + tok] = 0.0f;      // phase A: clear matched slot
    }
    __threadfence();
    __syncthreads();
    if (valid) {
        atomicAdd(&out[(size_t)b * V_ + tok], lp);  // phase B: accumulate
    }
}

// ---------------------------------------------------------------------------
extern "C" void kernel_launch(void* const* d_in, const int* in_sizes, int n_in,
                              void* d_out, int out_size, void* d_ws, size_t ws_size,
                              hipStream_t stream)
{
    const int*   hist     = (const int*)d_in[0];   // (S, B) int32
    const int*   idxp     = (const int*)d_in[1];   // scalar int32
    const int*   pointers = (const int*)d_in[2];   // (X,)  int32
    const int*   ids      = (const int*)d_in[3];   // (K,)  int32
    const float* logs     = (const float*)d_in[4]; // (L,)  float32
    float*       out      = (float*)d_out;         // (B, V) float32

    dim3 grid(V_ / TILE_V, B_);                    // (32, 32) blocks
    lm_base_fill<<<grid, TPB, 0, stream>>>(hist, idxp, logs, out);
    lm_scatter<<<1, B_ * C_, 0, stream>>>(hist, idxp, pointers, ids, logs, out);
}